// FastbatchcolorimageInterpNet_76312978915400
// MI455X (gfx1250) — compile-verified
//
#include <hip/hip_runtime.h>

typedef __attribute__((ext_vector_type(2))) float v2f;
typedef __attribute__((ext_vector_type(8))) float v8f;

// Problem constants (from reference setup_inputs)
#define BATCH 8
#define CHAN  3
#define NY    1024
#define NX    1024
#define PLANE (NY * NX)

#define NBLK  1024   // partial-sum blocks (multiple of 16 for WMMA stage)
#define NTHR  256    // 8 wave32 per block

// ---------------------------------------------------------------------------
// Stage 1: gather + per-block reduce.
// Each thread handles a strided set of points; accumulates 16 values
// (8 batches x {gc0, gc1}) summed over channels. Deterministic tree reduce.
// ---------------------------------------------------------------------------
__global__ __launch_bounds__(NTHR) void interp_gather_kernel(
    const float* __restrict__ imgs,
    const float* __restrict__ z,
    float* __restrict__ partials,
    int npts)
{
    float acc[2 * BATCH];
#pragma unroll
    for (int i = 0; i < 2 * BATCH; ++i) acc[i] = 0.0f;

    const int stride = NBLK * NTHR;
    for (int p = blockIdx.x * NTHR + threadIdx.x; p < npts; p += stride) {
        float zy = z[2 * p + 0];
        float zx = z[2 * p + 1];
        float x0y = zy * (float)(NY - 1);
        float x0x = zx * (float)(NX - 1);
        bool oob = (x0y < 0.0f) | (x0y > (float)(NY - 1)) |
                   (x0x < 0.0f) | (x0x > (float)(NX - 1));
        if (oob) { x0y = 0.0f; x0x = 0.0f; }
        // x0y, x0x >= 0 here, so int cast == floor
        int yg = (int)x0y;
        int xg = (int)x0x;
        // guard the +1 taps (JAX clamps OOB indices; z in [0,1) never hits this)
        yg = min(yg, NY - 2);
        xg = min(xg, NX - 2);
        float fy = (float)yg - x0y;   // in (-1, 0]
        float fx = (float)xg - x0x;
        float valid = oob ? 0.0f : 1.0f;

        const float* base = imgs + (yg * NX + xg);

#pragma unroll
        for (int b = 0; b < BATCH; ++b) {
            float sa1 = 0.0f, sa2 = 0.0f, sa3 = 0.0f;
#pragma unroll
            for (int c = 0; c < CHAN; ++c) {
                const float* pl = base + (b * CHAN + c) * PLANE;
                float g00 = pl[0];
                float g01 = pl[1];
                float g10 = pl[NX];
                float g11 = pl[NX + 1];
                sa1 += g10 - g00;
                sa2 += g11 - g01;
                sa3 += g01 - g00;
            }
            float d = sa1 - sa2;
            acc[2 * b + 0] += valid * (d * fx + sa1);
            acc[2 * b + 1] += valid * (d * fy + sa3);
        }
    }

    // ---- deterministic block reduction ----
    // wave32 xor-reduce: every lane ends with the full wave sum
#pragma unroll
    for (int i = 0; i < 2 * BATCH; ++i) {
        float v = acc[i];
#pragma unroll
        for (int off = 16; off > 0; off >>= 1)
            v += __shfl_xor(v, off, 32);
        acc[i] = v;
    }

    __shared__ float sred[NTHR / 32][2 * BATCH];
    const int wave = threadIdx.x >> 5;
    const int lane = threadIdx.x & 31;
    if (lane == 0) {
#pragma unroll
        for (int i = 0; i < 2 * BATCH; ++i) sred[wave][i] = acc[i];
    }
    __syncthreads();

    if (threadIdx.x < 2 * BATCH) {
        float s = 0.0f;
#pragma unroll
        for (int w = 0; w < NTHR / 32; ++w) s += sred[w][threadIdx.x];
        partials[blockIdx.x * (2 * BATCH) + threadIdx.x] = s;
    }
}

// ---------------------------------------------------------------------------
// Stage 2: reduce NBLK x 16 partials -> 16 outputs via V_WMMA_F32_16X16X4_F32.
// A = all-ones (16x4), B = 4 consecutive partial rows (4x16), so
// D[m][n] = sum_k B[k][n] + C[m][n]: every WMMA folds 4 rows into the column
// sums. 4 interleaved accumulator chains hide WMMA latency; deterministic
// fixed-order summation. One wave32, EXEC all ones (WMMA requirement).
// B-operand lane layout per ISA: VGPR0 = {lanes 0-15: K=0, lanes 16-31: K=2},
// VGPR1 = {K=1, K=3}, N = lane % 16.
// ---------------------------------------------------------------------------
__global__ __launch_bounds__(32) void reduce_wmma_kernel(
    const float* __restrict__ partials,
    float* __restrict__ out)
{
    const int lane = threadIdx.x;          // 0..31
    const int n    = lane & 15;            // column
    const int khi  = (lane >> 4) ? 2 : 0;  // K offset for this half-wave

    v2f a; a.x = 1.0f; a.y = 1.0f;         // ones matrix (all K, all M)
    v8f c0 = {}, c1 = {}, c2 = {}, c3 = {};

    for (int r = 0; r < NBLK; r += 16) {
        v2f b0, b1, b2, b3;
        b0.x = partials[(r +  0 + khi) * 16 + n];
        b0.y = partials[(r +  1 + khi) * 16 + n];
        b1.x = partials[(r +  4 + khi) * 16 + n];
        b1.y = partials[(r +  5 + khi) * 16 + n];
        b2.x = partials[(r +  8 + khi) * 16 + n];
        b2.y = partials[(r +  9 + khi) * 16 + n];
        b3.x = partials[(r + 12 + khi) * 16 + n];
        b3.y = partials[(r + 13 + khi) * 16 + n];
        c0 = __builtin_amdgcn_wmma_f32_16x16x4_f32(false, a, false, b0, (short)0, c0, false, false);
        c1 = __builtin_amdgcn_wmma_f32_16x16x4_f32(false, a, false, b1, (short)0, c1, false, false);
        c2 = __builtin_amdgcn_wmma_f32_16x16x4_f32(false, a, false, b2, (short)0, c2, false, false);
        c3 = __builtin_amdgcn_wmma_f32_16x16x4_f32(false, a, false, b3, (short)0, c3, false, false);
    }

    v8f c = c0 + c1 + c2 + c3;   // D layouts identical -> elementwise combine

    // C/D layout: VGPR0, lanes 0-15 hold row M=0, N=lane -> c[0] on lanes 0..15
    if (lane < 16) out[lane] = c[0];
}

// ---------------------------------------------------------------------------
extern "C" void kernel_launch(void* const* d_in, const int* in_sizes, int n_in,
                              void* d_out, int out_size, void* d_ws, size_t ws_size,
                              hipStream_t stream) {
    const float* imgs = (const float*)d_in[0];   // (8,3,1024,1024) f32
    const float* z    = (const float*)d_in[1];   // (N_PTS,2) f32
    float* partials   = (float*)d_ws;            // NBLK*16 floats = 64 KB
    float* out        = (float*)d_out;           // (8,2) f32

    const int npts = in_sizes[1] / 2;

    interp_gather_kernel<<<NBLK, NTHR, 0, stream>>>(imgs, z, partials, npts);
    reduce_wmma_kernel<<<1, 32, 0, stream>>>(partials, out);
}